// GridMAX2_11836929868046
// MI455X (gfx1250) — compile-verified
//
#include <hip/hip_runtime.h>
#include <stdint.h>

// cummax(x, axis=2), x: [8,128,128,512] fp32  ->  same shape.
// Pure streaming: 256MB in + 256MB out @ 23.3 TB/s => ~22us floor.
// Strategy: per (b,t) row, double-buffered async bulk copy of contiguous
// 32KB s-chunks into LDS (ASYNCcnt path), serial max chain in VGPRs,
// coalesced non-temporal float2 stores (output is written once, never read).

#define TS 128
#define CDIM 512
#define NROWS 1024                                  // B*Tt = 8*128
#define CHUNK_S 16
#define NCHUNK (TS / CHUNK_S)                       // 8
#define CHUNK_FLOATS (CHUNK_S * CDIM)               // 8192 floats = 32KB
#define THREADS 256
#define F4_PER_CHUNK (CHUNK_FLOATS / 4)             // 2048
#define LOADS_PER_THREAD (F4_PER_CHUNK / THREADS)   // 8 x b128 per thread

// Exact pointee type expected by the ROCm 7.2 builtin:
//   '__attribute__((__vector_size__(4 * sizeof(int)))) int __device__ *'
typedef int v4i __attribute__((vector_size(4 * sizeof(int))));
typedef __attribute__((address_space(1))) v4i* gv4i_ptr;
typedef __attribute__((address_space(3))) v4i* lv4i_ptr;

typedef float v2f __attribute__((ext_vector_type(2)));

__device__ __forceinline__ void async_copy_b128(const float* gsrc, float* ldst) {
#if __has_builtin(__builtin_amdgcn_global_load_async_to_lds_b128)
    __builtin_amdgcn_global_load_async_to_lds_b128(
        (gv4i_ptr)(uintptr_t)gsrc,
        (lv4i_ptr)(__attribute__((address_space(3))) void*)ldst,
        0, 0);
#else
    uint32_t laddr =
        (uint32_t)(uintptr_t)(__attribute__((address_space(3))) void*)ldst;
    asm volatile("global_load_async_to_lds_b128 %0, %1, off"
                 :: "v"(laddr), "v"((uint64_t)(uintptr_t)gsrc)
                 : "memory");
#endif
}

template <int N>
__device__ __forceinline__ void wait_asynccnt() {
#if __has_builtin(__builtin_amdgcn_s_wait_asynccnt)
    __builtin_amdgcn_s_wait_asynccnt(N);
#else
    asm volatile("s_wait_asynccnt %0" :: "i"(N) : "memory");
#endif
}

__global__ __launch_bounds__(THREADS)
void cummax_rows_kernel(const float* __restrict__ x, float* __restrict__ out) {
    __shared__ float lds[2][CHUNK_FLOATS];          // 64 KB, double buffer

    const int tid = threadIdx.x;
    const size_t row_base = (size_t)blockIdx.x * (TS * CDIM);
    const float* xrow = x + row_base;
    float* orow = out + row_base;

    // Prime the pipeline: chunk 0 -> buf 0 (contiguous 32KB, 16B per lane).
    #pragma unroll
    for (int i = 0; i < LOADS_PER_THREAD; ++i) {
        const int f4 = i * THREADS + tid;
        async_copy_b128(xrow + 4 * f4, &lds[0][4 * f4]);
    }

    float m0 = -__builtin_inff();
    float m1 = -__builtin_inff();
    const int c0 = 2 * tid;                          // this thread's 2 channels

    for (int k = 0; k < NCHUNK; ++k) {
        const int buf = k & 1;

        if (k + 1 < NCHUNK) {
            // Issue next chunk into the other buffer, then wait until only
            // those 8 are outstanding => chunk k has landed (in-order).
            const float* gsrc = xrow + (size_t)(k + 1) * CHUNK_FLOATS;
            float* lbase = &lds[buf ^ 1][0];
            #pragma unroll
            for (int i = 0; i < LOADS_PER_THREAD; ++i) {
                const int f4 = i * THREADS + tid;
                async_copy_b128(gsrc + 4 * f4, lbase + 4 * f4);
            }
            wait_asynccnt<LOADS_PER_THREAD>();
        } else {
            wait_asynccnt<0>();
        }
        __syncthreads();   // chunk k visible to all 8 waves

        // Serial prefix-max over 16 s-steps; ds_load_b64 reads are
        // bank-conflict-free (lanes cover all 64 banks in pairs).
        float* oc = orow + (size_t)k * CHUNK_FLOATS + c0;
        const float* lc = &lds[buf][c0];
        #pragma unroll
        for (int s = 0; s < CHUNK_S; ++s) {
            const v2f v = *(const v2f*)(lc + s * CDIM);
            m0 = fmaxf(m0, v.x);
            m1 = fmaxf(m1, v.y);
            v2f r = {m0, m1};
            // Output is streamed out exactly once: non-temporal store keeps
            // the 256MB write stream from evicting the inbound stream in L2.
            __builtin_nontemporal_store(r, (v2f*)(oc + s * CDIM));
        }
        __syncthreads();   // buf safe to overwrite by next iteration's issue
    }
}

extern "C" void kernel_launch(void* const* d_in, const int* in_sizes, int n_in,
                              void* d_out, int out_size, void* d_ws, size_t ws_size,
                              hipStream_t stream) {
    (void)in_sizes; (void)n_in; (void)out_size; (void)d_ws; (void)ws_size;
    const float* x = (const float*)d_in[0];
    float* out = (float*)d_out;
    cummax_rows_kernel<<<dim3(NROWS), dim3(THREADS), 0, stream>>>(x, out);
}